// UnitingModelForNLP_v1_9689446220129
// MI455X (gfx1250) — compile-verified
//
#include <hip/hip_runtime.h>
#include <math.h>

typedef float v2f __attribute__((ext_vector_type(2)));
typedef float v8f __attribute__((ext_vector_type(8)));

#define H_DIM 1024
#define B_DIM 64
#define L_DIM 3
#define S_DIM 512
#define NBINS 10

// ---------- ordered-int float mapping for atomic min/max ----------
__device__ __forceinline__ unsigned f2ord(float f) {
  unsigned u = __float_as_uint(f);
  return (u & 0x80000000u) ? ~u : (u | 0x80000000u);
}
__device__ __forceinline__ float ord2f(unsigned o) {
  unsigned u = (o & 0x80000000u) ? (o ^ 0x80000000u) : ~o;
  return __uint_as_float(u);
}

// ---------- init workspace counters (ws is poisoned 0xAA each timing run) ----
__global__ void k_init(unsigned* scratch) {
  int t = threadIdx.x;
  if (t < 4) scratch[t] = 0xFFFFFFFFu;       // mins (ordered space)
  else if (t < 48) scratch[t] = 0u;          // maxes [4..7], bins [8..47]
}

// ---------- per-matrix min/max reduction ----------
__global__ void k_minmax(const float* __restrict__ v, int n,
                         unsigned* mn, unsigned* mx) {
  __shared__ unsigned smn[256], smx[256];
  unsigned lmn = 0xFFFFFFFFu, lmx = 0u;
  for (int i = blockIdx.x * blockDim.x + threadIdx.x; i < n;
       i += gridDim.x * blockDim.x) {
    unsigned o = f2ord(v[i]);
    lmn = (o < lmn) ? o : lmn;
    lmx = (o > lmx) ? o : lmx;
  }
  smn[threadIdx.x] = lmn; smx[threadIdx.x] = lmx;
  __syncthreads();
  for (int s = 128; s > 0; s >>= 1) {
    if (threadIdx.x < s) {
      unsigned a = smn[threadIdx.x + s];
      unsigned b = smx[threadIdx.x + s];
      if (a < smn[threadIdx.x]) smn[threadIdx.x] = a;
      if (b > smx[threadIdx.x]) smx[threadIdx.x] = b;
    }
    __syncthreads();
  }
  if (threadIdx.x == 0) { atomicMin(mn, smn[0]); atomicMax(mx, smx[0]); }
}

// ---------- 10-bin histogram matching reference semantics ----------
__global__ void k_hist(const float* __restrict__ v, int n,
                       const unsigned* mn, const unsigned* mx, unsigned* bins) {
  __shared__ unsigned sb[NBINS];
  if (threadIdx.x < NBINS) sb[threadIdx.x] = 0u;
  __syncthreads();
  float lo = ord2f(*mn);
  float hi = ord2f(*mx);
  float scale = (hi - lo) / (float)NBINS;
  for (int i = blockIdx.x * blockDim.x + threadIdx.x; i < n;
       i += gridDim.x * blockDim.x) {
    float val = v[i];
#pragma unroll
    for (int b = 0; b < NBINS; b++) {
      float lower = lo + (float)b * scale;
      float upper = lower + scale;
      if (val >= lower && val < upper) atomicAdd(&sb[b], 1u);
    }
  }
  __syncthreads();
  if (threadIdx.x < NBINS && sb[threadIdx.x]) atomicAdd(&bins[threadIdx.x], sb[threadIdx.x]);
}

// ---------- entropies -> w_global scalars ----------
__global__ void k_finalize(const unsigned* scratch, float* wg) {
  if (threadIdx.x != 0 || blockIdx.x != 0) return;
  float ent[4];
  const float ns[4] = {(float)(H_DIM * H_DIM), (float)(H_DIM * H_DIM),
                       (float)(L_DIM * H_DIM), (float)(L_DIM * H_DIM)};
  for (int mi = 0; mi < 4; mi++) {
    float s = 0.0f;
    for (int b = 0; b < NBINS; b++) {
      float p = (float)scratch[8 + mi * NBINS + b] / ns[mi];
      if (p > 0.0f) s -= p * logf(p);
    }
    ent[mi] = s;
  }
  const float PI_F = 3.14159265358979323846f;
  wg[0] = 0.4f / PI_F * atanf(500.0f * (ent[0] - ent[1])) + 0.5f;
  wg[1] = 0.4f / PI_F * atanf(500.0f * (ent[2] - ent[3])) + 0.5f;
}

// ---------- dense unite: W = |pi-pj| @ wa^T, fused sigmoid/blend epilogue ----
// One wave32 per 16x64 strip (4 C tiles in N): A fragment (incl. |pi-pj|) is
// loaded once and reused for 4 WMMAs -> 6 loads per 4 v_wmma_f32_16x16x4_f32.
__global__ void __launch_bounds__(256) k_dense_unite(
    const float* __restrict__ pi, const float* __restrict__ pj,
    const float* __restrict__ wa, const float* __restrict__ ba,
    const float* __restrict__ wg, float* __restrict__ pu) {
  const int H = H_DIM;
  int wave = (blockIdx.x * blockDim.x + threadIdx.x) >> 5;
  int lane = threadIdx.x & 31;
  int tm = wave >> 4;            // 64 M tiles
  int tg = wave & 15;            // 16 N groups of 4 tiles (64 cols)
  int lanelo = lane & 15;
  int m = tm * 16 + lanelo;      // A-fragment row
  int khalf = (lane >> 4) << 1;  // lanes 0-15: k,k+1 ; lanes 16-31: k+2,k+3
  const float2* piRow = (const float2*)(pi + (size_t)m * H);
  const float2* pjRow = (const float2*)(pj + (size_t)m * H);
  const float2* waRow[4];
#pragma unroll
  for (int s = 0; s < 4; s++)
    waRow[s] = (const float2*)(wa + (size_t)(tg * 64 + s * 16 + lanelo) * H);

  v8f acc[4];
#pragma unroll
  for (int s = 0; s < 4; s++) acc[s] = (v8f){0.f,0.f,0.f,0.f,0.f,0.f,0.f,0.f};

  for (int kc = 0; kc < H; kc += 32) {
    if (kc + 32 < H) {  // pull next chunk toward the WGP (global_prefetch_b8)
      int kp = (kc + 32 + khalf) >> 1;
      __builtin_prefetch((const void*)(piRow + kp), 0, 1);
      __builtin_prefetch((const void*)(pjRow + kp), 0, 1);
#pragma unroll
      for (int s = 0; s < 4; s++)
        __builtin_prefetch((const void*)(waRow[s] + kp), 0, 1);
    }
#pragma unroll
    for (int k = kc; k < kc + 32; k += 4) {
      int k2 = (k + khalf) >> 1;
      float2 p2 = piRow[k2];
      float2 q2 = pjRow[k2];
      v2f a;
      a.x = fabsf(p2.x - q2.x);
      a.y = fabsf(p2.y - q2.y);      // fused |pi-pj|
#pragma unroll
      for (int s = 0; s < 4; s++) {
        float2 w2 = waRow[s][k2];    // B[k][n] = wa[n][k]
        v2f b; b.x = w2.x; b.y = w2.y;
        acc[s] = __builtin_amdgcn_wmma_f32_16x16x4_f32(
            false, a, false, b, (short)0, acc[s], false, false);
      }
    }
  }

  float wgv = wg[0];
#pragma unroll
  for (int s = 0; s < 4; s++) {
    int nn = tg * 64 + s * 16 + lanelo;
    float bias = ba[nn];
#pragma unroll
    for (int r = 0; r < 8; r++) {
      int mr = tm * 16 + r + ((lane >> 4) << 3);  // C/D: m = r + 8*(lane/16)
      float g = acc[s][r] + bias;
      float wl = 1.0f / (1.0f + expf(-g));        // sigmoid
      float w = wgv * wl + (1.0f - wgv);
      size_t idx = (size_t)mr * H + nn;
      pu[idx] = pi[idx] * w + pj[idx] * (1.0f - w);
    }
  }
}

// ---------- h = tanh(x[:,0,:] @ pu0^T + b), 16x64 strips ----------
__global__ void __launch_bounds__(256) k_hidden(
    const float* __restrict__ x, const float* __restrict__ pu,
    const float* __restrict__ bb, float* __restrict__ h) {
  const int H = H_DIM;
  int wave = (blockIdx.x * blockDim.x + threadIdx.x) >> 5;
  int lane = threadIdx.x & 31;
  int tm = wave >> 4;            // 4 M tiles (64 rows)
  int tg = wave & 15;            // 16 N groups of 4 tiles
  int lanelo = lane & 15;
  int m = tm * 16 + lanelo;
  int khalf = (lane >> 4) << 1;
  const float2* xRow = (const float2*)(x + (size_t)m * (size_t)(S_DIM * H_DIM)); // x[m,0,:]
  const float2* puRow[4];
#pragma unroll
  for (int s = 0; s < 4; s++)
    puRow[s] = (const float2*)(pu + (size_t)(tg * 64 + s * 16 + lanelo) * H);

  v8f acc[4];
#pragma unroll
  for (int s = 0; s < 4; s++) acc[s] = (v8f){0.f,0.f,0.f,0.f,0.f,0.f,0.f,0.f};

  for (int kc = 0; kc < H; kc += 32) {
    if (kc + 32 < H) {
      int kp = (kc + 32 + khalf) >> 1;
      __builtin_prefetch((const void*)(xRow + kp), 0, 1);
#pragma unroll
      for (int s = 0; s < 4; s++)
        __builtin_prefetch((const void*)(puRow[s] + kp), 0, 1);
    }
#pragma unroll
    for (int k = kc; k < kc + 32; k += 4) {
      int k2 = (k + khalf) >> 1;
      float2 a2 = xRow[k2];
      v2f a; a.x = a2.x; a.y = a2.y;
#pragma unroll
      for (int s = 0; s < 4; s++) {
        float2 b2 = puRow[s][k2];
        v2f b; b.x = b2.x; b.y = b2.y;
        acc[s] = __builtin_amdgcn_wmma_f32_16x16x4_f32(
            false, a, false, b, (short)0, acc[s], false, false);
      }
    }
  }

#pragma unroll
  for (int s = 0; s < 4; s++) {
    int nn = tg * 64 + s * 16 + lanelo;
    float bias = bb[nn];
#pragma unroll
    for (int r = 0; r < 8; r++) {
      int mr = tm * 16 + r + ((lane >> 4) << 3);
      h[(size_t)mr * H + nn] = tanhf(acc[s][r] + bias);
    }
  }
}

// ---------- pu1 = unite(base_out_w, graft_out_w, ...) : 3x1024, scalar -------
__global__ void k_out_unite(const float* __restrict__ bo, const float* __restrict__ go,
                            const float* __restrict__ oa, const float* __restrict__ ob,
                            const float* __restrict__ wg, float* __restrict__ pu1) {
  int t = blockIdx.x * blockDim.x + threadIdx.x;
  if (t >= L_DIM * H_DIM) return;
  int r = t / H_DIM, n = t % H_DIM;
  const float* br = bo + (size_t)r * H_DIM;
  const float* gr = go + (size_t)r * H_DIM;
  const float* ar = oa + (size_t)n * H_DIM;
  float g = 0.0f;
  for (int k = 0; k < H_DIM; k++) g += fabsf(br[k] - gr[k]) * ar[k];
  g += ob[n];
  float wl = 1.0f / (1.0f + expf(-g));
  float wgv = wg[1];
  float w = wgv * wl + (1.0f - wgv);
  pu1[t] = bo[t] * w + go[t] * (1.0f - w);
}

// ---------- out = h @ pu1^T + base_out_b : 64x3 ----------
__global__ void k_final(const float* __restrict__ h, const float* __restrict__ pu1,
                        const float* __restrict__ bob, float* __restrict__ out) {
  int t = blockIdx.x * blockDim.x + threadIdx.x;
  if (t >= B_DIM * L_DIM) return;
  int m = t / L_DIM, l = t % L_DIM;
  const float* hr = h + (size_t)m * H_DIM;
  const float* pr = pu1 + (size_t)l * H_DIM;
  float s = 0.0f;
  for (int k = 0; k < H_DIM; k++) s += hr[k] * pr[k];
  out[t] = s + bob[l];
}

extern "C" void kernel_launch(void* const* d_in, const int* in_sizes, int n_in,
                              void* d_out, int out_size, void* d_ws, size_t ws_size,
                              hipStream_t stream) {
  const float* x   = (const float*)d_in[0];   // (64,512,1024)
  const float* bdw = (const float*)d_in[1];   // (1024,1024)
  const float* bdb = (const float*)d_in[2];   // (1024,)
  const float* bow = (const float*)d_in[3];   // (3,1024)
  const float* bob = (const float*)d_in[4];   // (3,)
  const float* gdw = (const float*)d_in[5];   // (1024,1024)
  const float* gow = (const float*)d_in[6];   // (3,1024)
  const float* daw = (const float*)d_in[7];   // (1024,1024)
  const float* dab = (const float*)d_in[8];   // (1024,)
  const float* oaw = (const float*)d_in[9];   // (1024,1024)
  const float* oab = (const float*)d_in[10];  // (1024,)
  float* out = (float*)d_out;                 // (64,3)

  // workspace layout: [0,256) u32 counters | [256,512) wg scalars |
  // pu0 4MiB | h 256KiB | pu1 12KiB  (~4.26 MiB total)
  unsigned* scratch = (unsigned*)d_ws;
  float* wg  = (float*)((char*)d_ws + 256);
  float* pu0 = (float*)((char*)d_ws + 512);
  float* h   = pu0 + (size_t)H_DIM * H_DIM;
  float* pu1 = h + (size_t)B_DIM * H_DIM;

  const int nBig = H_DIM * H_DIM, nSmall = L_DIM * H_DIM;

  k_init<<<1, 64, 0, stream>>>(scratch);
  k_minmax<<<256, 256, 0, stream>>>(bdw, nBig,   scratch + 0, scratch + 4);
  k_minmax<<<256, 256, 0, stream>>>(gdw, nBig,   scratch + 1, scratch + 5);
  k_minmax<<<8,   256, 0, stream>>>(bow, nSmall, scratch + 2, scratch + 6);
  k_minmax<<<8,   256, 0, stream>>>(gow, nSmall, scratch + 3, scratch + 7);
  k_hist<<<256, 256, 0, stream>>>(bdw, nBig,   scratch + 0, scratch + 4, scratch + 8);
  k_hist<<<256, 256, 0, stream>>>(gdw, nBig,   scratch + 1, scratch + 5, scratch + 18);
  k_hist<<<8,   256, 0, stream>>>(bow, nSmall, scratch + 2, scratch + 6, scratch + 28);
  k_hist<<<8,   256, 0, stream>>>(gow, nSmall, scratch + 3, scratch + 7, scratch + 38);
  k_finalize<<<1, 1, 0, stream>>>(scratch, wg);

  // 1024^3 FP32 GEMM: 64 M-tiles x 16 N-groups, 1 wave per 16x64 strip
  k_dense_unite<<<128, 256, 0, stream>>>(bdw, gdw, daw, dab, wg, pu0);
  // 64x1024x1024 GEMM: 4 M-tiles x 16 N-groups
  k_hidden<<<8, 256, 0, stream>>>(x, pu0, bdb, h);
  k_out_unite<<<(L_DIM * H_DIM + 255) / 256, 256, 0, stream>>>(bow, gow, oaw, oab, wg, pu1);
  k_final<<<1, 192, 0, stream>>>(h, pu1, bob, out);
}